// FKLossB_83348135346475
// MI455X (gfx1250) — compile-verified
//
#include <hip/hip_runtime.h>
#include <math.h>

typedef float v2f __attribute__((ext_vector_type(2)));
typedef float v8f __attribute__((ext_vector_type(8)));

// Full-wave (32 lane) f32 sum using the CDNA5 f32 WMMA.
// A = 16x4 with lane partials at K=0 (lanes 0-15) / K=2 (lanes 16-31), B = ones.
// D[m][n] = partial_m + partial_{16+m}; per-lane sum of the 8 D VGPRs gives the
// half-sums, one xor-shuffle completes the 32-lane total on every lane.
__device__ __forceinline__ float wave_reduce_wmma(float partial) {
    v2f a; a.x = partial; a.y = 0.0f;
    v2f b; b.x = 1.0f;    b.y = 1.0f;
    v8f c = {};
    c = __builtin_amdgcn_wmma_f32_16x16x4_f32(
            /*neg_a=*/false, a, /*neg_b=*/false, b,
            /*c_mod=*/(short)0, c, /*reuse_a=*/false, /*reuse_b=*/false);
    float s = c.s0 + c.s1 + c.s2 + c.s3 + c.s4 + c.s5 + c.s6 + c.s7;
    s += __shfl_xor(s, 16, 32);   // wave32: combine rows 0-7 half with rows 8-15 half
    return s;
}

// float32 cos(pi/2) exactly as JAX/f32 sees it (pi/2 rounds up by ~4.37e-8)
#define CAL_HALF_PI (-4.371139e-08f)

__device__ __forceinline__ float fk_err(const float* __restrict__ j,
                                        const float* __restrict__ p) {
    const float dD [7] = {0.333f, 0.0f, 0.316f, 0.0f,    0.384f,   0.0f, 0.107f};
    const float aA [7] = {0.0f,   0.0f, 0.0f,   0.0825f, -0.0825f, 0.0f, 0.088f};
    const float cal[7] = {1.0f, CAL_HALF_PI, CAL_HALF_PI, CAL_HALF_PI,
                          CAL_HALF_PI, CAL_HALF_PI, CAL_HALF_PI};
    const float sal[7] = {0.0f, -1.0f, 1.0f, 1.0f, -1.0f, 1.0f, 1.0f};

    // T as 3x4 affine, start at identity
    float R00=1.f,R01=0.f,R02=0.f, R10=0.f,R11=1.f,R12=0.f, R20=0.f,R21=0.f,R22=1.f;
    float tx=0.f, ty=0.f, tz=0.f;
#pragma unroll
    for (int i = 0; i < 7; ++i) {
        float st, ct;
        sincosf(j[i], &st, &ct);
        const float ca = cal[i], sa = sal[i], a = aA[i], d = dD[i];
        // columns of A_i (row3 = [0 0 0 1] dropped)
        const float b00 =  ct,      b10 =  st;               // col0 (z comp = 0)
        const float b01 = -st * ca, b11 =  ct * ca, b21 = sa; // col1
        const float b02 =  st * sa, b12 = -ct * sa, b22 = ca; // col2
        const float b03 =  a * ct,  b13 =  a * st,  b23 = d;  // translation
        const float n00 = R00*b00 + R01*b10;
        const float n10 = R10*b00 + R11*b10;
        const float n20 = R20*b00 + R21*b10;
        const float n01 = R00*b01 + R01*b11 + R02*b21;
        const float n11 = R10*b01 + R11*b11 + R12*b21;
        const float n21 = R20*b01 + R21*b11 + R22*b21;
        const float n02 = R00*b02 + R01*b12 + R02*b22;
        const float n12 = R10*b02 + R11*b12 + R12*b22;
        const float n22 = R20*b02 + R21*b12 + R22*b22;
        tx += R00*b03 + R01*b13 + R02*b23;
        ty += R10*b03 + R11*b13 + R12*b23;
        tz += R20*b03 + R21*b13 + R22*b23;
        R00=n00; R01=n01; R02=n02;
        R10=n10; R11=n11; R12=n12;
        R20=n20; R21=n21; R22=n22;
    }
    const float roll  = atan2f(-R12, R22);
    const float pitch = asinf(fminf(1.0f, fmaxf(-1.0f, R02)));
    const float yaw   = atan2f(-R01, R00);

    const float e0 = tx    - p[0];
    const float e1 = ty    - p[1];
    const float e2 = tz    - p[2];
    const float e3 = roll  - p[3];
    const float e4 = pitch - p[4];
    const float e5 = yaw   - p[5];
    return e0*e0 + e1*e1 + e2*e2 + e3*e3 + e4*e4 + e5*e5;
}

__global__ __launch_bounds__(256)
void fk_loss_partial(const float* __restrict__ joints,
                     const float* __restrict__ poses,
                     float* __restrict__ blockSums, int B) {
    const int b = blockIdx.x * blockDim.x + threadIdx.x;
    float e = 0.0f;
    if (b < B)  // reconverges before the WMMA reduce (EXEC all-ones there)
        e = fk_err(joints + (size_t)b * 7, poses + (size_t)b * 6);

    const float wsum = wave_reduce_wmma(e);

    __shared__ float lds[8];
    const int wave = threadIdx.x >> 5;
    const int lane = threadIdx.x & 31;
    if (lane == 0) lds[wave] = wsum;
    __syncthreads();
    if (threadIdx.x == 0) {
        float s = 0.0f;
        const int nw = blockDim.x >> 5;
        for (int i = 0; i < nw; ++i) s += lds[i];   // fixed order -> deterministic
        blockSums[blockIdx.x] = s;
    }
}

__global__ __launch_bounds__(32)
void fk_loss_final(const float* __restrict__ blockSums, int nblocks,
                   float* __restrict__ out, float invCount) {
    const int lane = threadIdx.x;
    float s = 0.0f;
    for (int i = lane; i < nblocks; i += 32) s += blockSums[i];  // fixed order
    const float total = wave_reduce_wmma(s);
    if (lane == 0) out[0] = total * invCount;
}

extern "C" void kernel_launch(void* const* d_in, const int* in_sizes, int n_in,
                              void* d_out, int out_size, void* d_ws, size_t ws_size,
                              hipStream_t stream) {
    const float* joints = (const float*)d_in[0];   // (B,7) f32
    const float* poses  = (const float*)d_in[1];   // (B,6) f32
    const int B = in_sizes[0] / 7;

    float* blockSums = (float*)d_ws;               // nblocks floats of scratch
    const int threads = 256;
    const int nblocks = (B + threads - 1) / threads;

    fk_loss_partial<<<nblocks, threads, 0, stream>>>(joints, poses, blockSums, B);

    const float invCount = 1.0f / (6.0f * (float)B);
    fk_loss_final<<<1, 32, 0, stream>>>(blockSums, nblocks, (float*)d_out, invCount);
}